// GAGNN_65747359367502
// MI455X (gfx1250) — compile-verified
//
#include <hip/hip_runtime.h>
#include <hip/hip_bf16.h>

// ---------------------------------------------------------------------------
// GAGNN on MI455X (gfx1250): fp32 WMMA (V_WMMA_F32_16X16X4_F32) for all GEMMs.
// B matrices stored in LDS interleaved by k-pairs (float2) so each WMMA B
// fragment is a single ds_load_b64 into an aligned VGPR pair (no repacking).
// ---------------------------------------------------------------------------

typedef float v2f __attribute__((ext_vector_type(2)));
typedef float v8f __attribute__((ext_vector_type(8)));

#define M_TOT   16000   // B*N
#define T_LEN   24
#define F_IN    10
#define HID     128
#define G3      384     // 3*HID
#define G3P2    400     // padded v2f row stride (dword stride 800 ≡ 32 mod 64)
#define E_ONE   8000
#define E_TOT   256000  // B*E
#define CHUNK   3200    // rows per GRU chunk (5 chunks); h1/xp1 stay ~L2-sized

__device__ __forceinline__ v8f wmma4(v2f a, v2f b, v8f c) {
    // D = A(16x4,f32) * B(4x16,f32) + C(16x16,f32)
    return __builtin_amdgcn_wmma_f32_16x16x4_f32(false, a, false, b,
                                                 (short)0, c, false, false);
}
__device__ __forceinline__ float sigm(float v) { return 1.f / (1.f + __expf(-v)); }

__device__ __forceinline__ void atomicMaxF(float* a, float v) {
    if (v >= 0.f) atomicMax((int*)a, __float_as_int(v));
    else          atomicMin((unsigned int*)a, __float_as_uint(v));
}

// ---------------------------------------------------------------------------
// GRU layer 0 scan: x(M,T,10) -> h1 chunk (CHUNK*T,128).  4 waves / block,
// 64 rows / block.  Whh0^T + Wih0^T (K padded to 12) resident in LDS.
// ---------------------------------------------------------------------------
__global__ void __launch_bounds__(128)
gru0_scan(const float* __restrict__ x, const float* __restrict__ Wih,
          const float* __restrict__ Whh, const float* __restrict__ bih,
          const float* __restrict__ bhh, float* __restrict__ h1c, int chunk0)
{
    __shared__ v2f   s_whh2[64 * G3P2];             // {WhhT[2kp][g], WhhT[2kp+1][g]}
    __shared__ v2f   s_wih2[6 * G3P2];              // k padded 10 -> 12
    __shared__ __attribute__((aligned(16))) float s_h [4 * 16 * 132];
    __shared__ __attribute__((aligned(16))) float s_xn[4 * 16 * 132];
    __shared__ float s_bih[G3], s_bhh[G3];

    const int tid = threadIdx.x;
    for (int i = tid; i < 64 * G3; i += 128) {
        int kp = i / G3, g = i % G3;
        v2f v; v.x = Whh[g * HID + 2 * kp]; v.y = Whh[g * HID + 2 * kp + 1];
        s_whh2[kp * G3P2 + g] = v;
    }
    for (int i = tid; i < 6 * G3; i += 128) {
        int kp = i / G3, g = i % G3;
        v2f v;
        v.x = (2 * kp     < F_IN) ? Wih[g * F_IN + 2 * kp]     : 0.f;
        v.y = (2 * kp + 1 < F_IN) ? Wih[g * F_IN + 2 * kp + 1] : 0.f;
        s_wih2[kp * G3P2 + g] = v;
    }
    for (int i = tid; i < G3; i += 128) { s_bih[i] = bih[i]; s_bhh[i] = bhh[i]; }
    for (int i = tid; i < 4 * 16 * 132; i += 128) s_h[i] = 0.f;
    __syncthreads();

    const int w = tid >> 5, lane = tid & 31, lo = lane & 15, hi = lane >> 4;
    const int lmBase = blockIdx.x * 64 + w * 16;
    const long mrow  = (long)(chunk0 + lmBase + lo);
    float* sh = &s_h[w * 16 * 132];
    float* sx = &s_xn[w * 16 * 132];

    for (int t = 0; t < T_LEN; ++t) {
        // A fragments of x_t (K padded 10 -> 12)
        v2f afr[3];
        #pragma unroll
        for (int kk = 0; kk < 3; ++kk) {
            int k0 = kk * 4 + hi * 2;
            afr[kk].x = (k0     < F_IN) ? x[mrow * (T_LEN * F_IN) + t * F_IN + k0]     : 0.f;
            afr[kk].y = (k0 + 1 < F_IN) ? x[mrow * (T_LEN * F_IN) + t * F_IN + k0 + 1] : 0.f;
        }
        // ---- n-gate input projection (cols 256..383), staged to LDS ----
        {
            v8f xacc[8];
            #pragma unroll
            for (int nt = 0; nt < 8; ++nt) {
                float bv = s_bih[256 + nt * 16 + lo];
                #pragma unroll
                for (int j = 0; j < 8; ++j) xacc[nt][j] = bv;
            }
            #pragma unroll
            for (int kk = 0; kk < 3; ++kk) {
                int kp = kk * 2 + hi;
                #pragma unroll
                for (int nt = 0; nt < 8; ++nt) {
                    v2f b = s_wih2[kp * G3P2 + 256 + nt * 16 + lo];
                    xacc[nt] = wmma4(afr[kk], b, xacc[nt]);
                }
            }
            #pragma unroll
            for (int nt = 0; nt < 8; ++nt)
                #pragma unroll
                for (int j = 0; j < 8; ++j)
                    sx[(j + hi * 8) * 132 + nt * 16 + lo] = xacc[nt][j];
        }
        // ---- r,z gates (merged x+h) and n-gate recurrent part ----
        v8f acc[24];
        #pragma unroll
        for (int nt = 0; nt < 16; ++nt) {
            float bv = s_bih[nt * 16 + lo] + s_bhh[nt * 16 + lo];
            #pragma unroll
            for (int j = 0; j < 8; ++j) acc[nt][j] = bv;
        }
        #pragma unroll
        for (int nt = 16; nt < 24; ++nt) {
            float bv = s_bhh[nt * 16 + lo];
            #pragma unroll
            for (int j = 0; j < 8; ++j) acc[nt][j] = bv;
        }
        #pragma unroll
        for (int kk = 0; kk < 3; ++kk) {
            int kp = kk * 2 + hi;
            #pragma unroll
            for (int nt = 0; nt < 16; ++nt) {
                v2f b = s_wih2[kp * G3P2 + nt * 16 + lo];
                acc[nt] = wmma4(afr[kk], b, acc[nt]);
            }
        }
        for (int kk = 0; kk < 32; ++kk) {
            int k0 = kk * 4 + hi * 2;
            v2f a = *(const v2f*)&sh[lo * 132 + k0];
            int kp = kk * 2 + hi;
            #pragma unroll
            for (int nt = 0; nt < 24; ++nt) {
                v2f b = s_whh2[kp * G3P2 + nt * 16 + lo];
                acc[nt] = wmma4(a, b, acc[nt]);
            }
        }
        // ---- gates (pure register VALU thanks to identical D layouts) ----
        #pragma unroll
        for (int nt = 0; nt < 8; ++nt) {
            #pragma unroll
            for (int j = 0; j < 8; ++j) {
                int row = j + hi * 8, col = nt * 16 + lo;
                float r  = sigm(acc[nt][j]);
                float z  = sigm(acc[nt + 8][j]);
                float hn = acc[nt + 16][j];
                float xn = sx[row * 132 + col];
                float nn = tanhf(xn + r * hn);
                float ho = sh[row * 132 + col];
                float hv = (1.f - z) * nn + z * ho;
                sh[row * 132 + col] = hv;
                h1c[((long)(lmBase + row) * T_LEN + t) * HID + col] = hv;
            }
        }
        __syncthreads();
    }
}

// ---------------------------------------------------------------------------
// Generic K=128 WMMA GEMM: C(R x NC) = A(R x 128) @ Wt^T (+ bias).
// Wt is (NC x 128) row-major. B kept k-pair interleaved in LDS.
// ---------------------------------------------------------------------------
template <int NT, int WR, int NC, bool BIAS>
__global__ void __launch_bounds__(128)
gemm_h128(const float* __restrict__ A, const float* __restrict__ Wt,
          const float* __restrict__ bias, float* __restrict__ C)
{
    constexpr int NCP2 = NC + 16;        // v2f stride; dword stride ≡ 32 mod 64
    __shared__ v2f s_b2[64 * NCP2];
    __shared__ __attribute__((aligned(16))) float s_a[WR * 16 * 132];

    const int tid = threadIdx.x;
    for (int i = tid; i < 64 * NC; i += 128) {
        int kp = i / NC, n = i % NC;
        v2f v; v.x = Wt[n * HID + 2 * kp]; v.y = Wt[n * HID + 2 * kp + 1];
        s_b2[kp * NCP2 + n] = v;
    }
    const int w = tid >> 5, lane = tid & 31, lo = lane & 15, hi = lane >> 4;
    const int rt = w % WR, cg = w / WR;
    const int rowBase = blockIdx.x * (WR * 16) + rt * 16;
    if (cg == 0) {
        for (int i = lane; i < 16 * HID; i += 32) {
            int r = i >> 7, c2 = i & 127;
            s_a[rt * 16 * 132 + r * 132 + c2] = A[(long)(rowBase + r) * HID + c2];
        }
    }
    __syncthreads();

    const int colBase = cg * (NT * 16);
    v8f acc[NT];
    #pragma unroll
    for (int nt = 0; nt < NT; ++nt) {
        float bv = 0.f;
        if constexpr (BIAS) bv = bias[colBase + nt * 16 + lo];
        #pragma unroll
        for (int j = 0; j < 8; ++j) acc[nt][j] = bv;
    }
    for (int kk = 0; kk < 32; ++kk) {
        int k0 = kk * 4 + hi * 2, kp = kk * 2 + hi;
        v2f a = *(const v2f*)&s_a[rt * 16 * 132 + lo * 132 + k0];
        #pragma unroll
        for (int nt = 0; nt < NT; ++nt) {
            v2f b = s_b2[kp * NCP2 + colBase + nt * 16 + lo];
            acc[nt] = wmma4(a, b, acc[nt]);
        }
    }
    #pragma unroll
    for (int nt = 0; nt < NT; ++nt)
        #pragma unroll
        for (int j = 0; j < 8; ++j)
            C[(long)(rowBase + j + hi * 8) * NC + colBase + nt * 16 + lo] = acc[nt][j];
}

// ---------------------------------------------------------------------------
// GRU layer 1 scan: xp1 chunk precomputed; Whh1^T resident in LDS.
// Writes only the final hidden state h2[:, -1, :] into h_node.
// ---------------------------------------------------------------------------
__global__ void __launch_bounds__(128)
gru1_scan(const float* __restrict__ xp, const float* __restrict__ Whh,
          const float* __restrict__ bhh, float* __restrict__ hout, int chunk0)
{
    __shared__ v2f s_whh2[64 * G3P2];
    __shared__ __attribute__((aligned(16))) float s_h[4 * 16 * 132];
    __shared__ float s_bhh[G3];

    const int tid = threadIdx.x;
    for (int i = tid; i < 64 * G3; i += 128) {
        int kp = i / G3, g = i % G3;
        v2f v; v.x = Whh[g * HID + 2 * kp]; v.y = Whh[g * HID + 2 * kp + 1];
        s_whh2[kp * G3P2 + g] = v;
    }
    for (int i = tid; i < G3; i += 128) s_bhh[i] = bhh[i];
    for (int i = tid; i < 4 * 16 * 132; i += 128) s_h[i] = 0.f;
    __syncthreads();

    const int w = tid >> 5, lane = tid & 31, lo = lane & 15, hi = lane >> 4;
    const int lmBase = blockIdx.x * 64 + w * 16;
    float* sh = &s_h[w * 16 * 132];

    for (int t = 0; t < T_LEN; ++t) {
        v8f acc[24];
        #pragma unroll
        for (int nt = 0; nt < 16; ++nt) {
            float bv = s_bhh[nt * 16 + lo];
            #pragma unroll
            for (int j = 0; j < 8; ++j)
                acc[nt][j] = xp[((long)(lmBase + j + hi * 8) * T_LEN + t) * G3 + nt * 16 + lo] + bv;
        }
        #pragma unroll
        for (int nt = 16; nt < 24; ++nt) {
            float bv = s_bhh[nt * 16 + lo];
            #pragma unroll
            for (int j = 0; j < 8; ++j) acc[nt][j] = bv;
        }
        for (int kk = 0; kk < 32; ++kk) {
            int k0 = kk * 4 + hi * 2, kp = kk * 2 + hi;
            v2f a = *(const v2f*)&sh[lo * 132 + k0];
            #pragma unroll
            for (int nt = 0; nt < 24; ++nt) {
                v2f b = s_whh2[kp * G3P2 + nt * 16 + lo];
                acc[nt] = wmma4(a, b, acc[nt]);
            }
        }
        #pragma unroll
        for (int nt = 0; nt < 8; ++nt) {
            #pragma unroll
            for (int j = 0; j < 8; ++j) {
                int row = j + hi * 8, col = nt * 16 + lo;
                float r  = sigm(acc[nt][j]);
                float z  = sigm(acc[nt + 8][j]);
                float hn = acc[nt + 16][j];
                float xn = xp[((long)(lmBase + row) * T_LEN + t) * G3 + 256 + col];
                float nn = tanhf(xn + r * hn);
                float ho = sh[row * 132 + col];
                float hv = (1.f - z) * nn + z * ho;
                sh[row * 132 + col] = hv;
                if (t == T_LEN - 1)
                    hout[(long)(chunk0 + lmBase + row) * HID + col] = hv;
            }
        }
        __syncthreads();
    }
}

// ---------------------------------------------------------------------------
// GAT helper kernels
// ---------------------------------------------------------------------------
__global__ void gat_init(float* segm, float* segs, float* agg)
{
    long i = (long)blockIdx.x * 256 + threadIdx.x;
    if (i < M_TOT * 4) { segm[i] = -INFINITY; segs[i] = 0.f; }
    if (i < (long)M_TOT * HID) agg[i] = 0.f;
}

__global__ void att_kernel(const float* __restrict__ xh, const float* __restrict__ asrc,
                           const float* __restrict__ adst, float* __restrict__ a_s,
                           float* __restrict__ a_d)
{
    int i = blockIdx.x * 256 + threadIdx.x;   // i < M*4
    int m = i >> 2, hd = i & 3;
    float ss = 0.f, sd = 0.f;
    for (int f = 0; f < HID; ++f) {
        float v = xh[(long)m * 512 + hd * HID + f];
        ss += v * asrc[hd * HID + f];
        sd += v * adst[hd * HID + f];
    }
    a_s[i] = ss; a_d[i] = sd;
}

__global__ void edge_max(const int* __restrict__ ei, const float* __restrict__ a_s,
                         const float* __restrict__ a_d, float* __restrict__ segm)
{
    int idx = blockIdx.x * 256 + threadIdx.x;   // < E_TOT
    int b = idx / E_ONE, e0 = idx - b * E_ONE;
    int src = ei[e0] + b * 500, dst = ei[E_ONE + e0] + b * 500;
    #pragma unroll
    for (int hd = 0; hd < 4; ++hd) {
        float e = a_s[src * 4 + hd] + a_d[dst * 4 + hd];
        e = (e > 0.f) ? e : 0.2f * e;
        atomicMaxF(&segm[dst * 4 + hd], e);
    }
}

__global__ void edge_exp(const int* __restrict__ ei, const float* __restrict__ a_s,
                         const float* __restrict__ a_d, const float* __restrict__ segm,
                         float* __restrict__ segs, float* __restrict__ exbuf)
{
    int idx = blockIdx.x * 256 + threadIdx.x;
    int b = idx / E_ONE, e0 = idx - b * E_ONE;
    int src = ei[e0] + b * 500, dst = ei[E_ONE + e0] + b * 500;
    #pragma unroll
    for (int hd = 0; hd < 4; ++hd) {
        float e = a_s[src * 4 + hd] + a_d[dst * 4 + hd];
        e = (e > 0.f) ? e : 0.2f * e;
        float ex = __expf(e - segm[dst * 4 + hd]);
        exbuf[(long)idx * 4 + hd] = ex;
        atomicAdd(&segs[dst * 4 + hd], ex);
    }
}

// one wave per edge; the 4-head mean is folded in-register => 4x fewer atomics
__global__ void edge_agg(const int* __restrict__ ei, const float* __restrict__ exbuf,
                         const float* __restrict__ segs, const float* __restrict__ xh,
                         float* __restrict__ agg)
{
    int w = threadIdx.x >> 5, lane = threadIdx.x & 31;
    long e = (long)blockIdx.x * 8 + w;          // 32000 blocks * 8 waves = E_TOT
    int b = (int)(e / E_ONE), e0 = (int)(e - (long)b * E_ONE);
    int src = ei[e0] + b * 500, dst = ei[E_ONE + e0] + b * 500;
    float al[4];
    #pragma unroll
    for (int hd = 0; hd < 4; ++hd)
        al[hd] = exbuf[e * 4 + hd] / (segs[dst * 4 + hd] + 1e-16f);
    #pragma unroll
    for (int i = 0; i < 4; ++i) {
        int f = i * 32 + lane;
        float v = 0.f;
        #pragma unroll
        for (int hd = 0; hd < 4; ++hd)
            v += al[hd] * xh[(long)src * 512 + hd * HID + f];
        atomicAdd(&agg[(long)dst * HID + f], 0.25f * v);
    }
}

// h = LayerNorm(relu(agg + gat_b) + h); one wave per node, shuffle reductions
__global__ void relu_res_ln(const float* __restrict__ agg, const float* __restrict__ gb,
                            const float* __restrict__ g, const float* __restrict__ bb,
                            float* __restrict__ h)
{
    int w = threadIdx.x >> 5, lane = threadIdx.x & 31;
    int m = blockIdx.x * 8 + w;
    float v[4];
    float s = 0.f;
    #pragma unroll
    for (int i = 0; i < 4; ++i) {
        int f = i * 32 + lane;
        float t = agg[(long)m * HID + f] + gb[f];
        t = (t > 0.f) ? t : 0.f;
        t += h[(long)m * HID + f];
        v[i] = t; s += t;
    }
    for (int off = 16; off; off >>= 1) s += __shfl_xor(s, off, 32);
    float mu = s * (1.f / HID);
    float q = 0.f;
    #pragma unroll
    for (int i = 0; i < 4; ++i) { float d = v[i] - mu; q += d * d; }
    for (int off = 16; off; off >>= 1) q += __shfl_xor(q, off, 32);
    float inv = rsqrtf(q * (1.f / HID) + 1e-5f);
    #pragma unroll
    for (int i = 0; i < 4; ++i) {
        int f = i * 32 + lane;
        h[(long)m * HID + f] = (v[i] - mu) * inv * g[f] + bb[f];
    }
}

__global__ void fc_kernel(const float* __restrict__ h, const float* __restrict__ Wf,
                          const float* __restrict__ bf, float* __restrict__ out)
{
    int i = blockIdx.x * 256 + threadIdx.x;   // < M*6
    int m = i / 6, j = i - m * 6;
    float s = bf[j];
    for (int k = 0; k < HID; ++k) s += h[(long)m * HID + k] * Wf[j * HID + k];
    out[i] = s;
}

// ---------------------------------------------------------------------------
extern "C" void kernel_launch(void* const* d_in, const int* in_sizes, int n_in,
                              void* d_out, int out_size, void* d_ws, size_t ws_size,
                              hipStream_t stream)
{
    const float* x     = (const float*)d_in[0];
    const float* Wih0  = (const float*)d_in[1];
    const float* Whh0  = (const float*)d_in[2];
    const float* bih0  = (const float*)d_in[3];
    const float* bhh0  = (const float*)d_in[4];
    const float* Wih1  = (const float*)d_in[5];
    const float* Whh1  = (const float*)d_in[6];
    const float* bih1  = (const float*)d_in[7];
    const float* bhh1  = (const float*)d_in[8];
    const float* gW[2]    = { (const float*)d_in[9],  (const float*)d_in[13] };
    const float* gAsrc[2] = { (const float*)d_in[10], (const float*)d_in[14] };
    const float* gAdst[2] = { (const float*)d_in[11], (const float*)d_in[15] };
    const float* gB[2]    = { (const float*)d_in[12], (const float*)d_in[16] };
    const float* lnG[2]   = { (const float*)d_in[17], (const float*)d_in[19] };
    const float* lnB[2]   = { (const float*)d_in[18], (const float*)d_in[20] };
    const float* fcW   = (const float*)d_in[21];
    const float* fcB   = (const float*)d_in[22];
    const int*   ei    = (const int*)d_in[23];
    float* out = (float*)d_out;

    char* p = (char*)d_ws;
    auto alloc = [&](size_t nflt) {
        float* r = (float*)p;
        p += ((nflt * sizeof(float) + 255) / 256) * 256;
        return r;
    };
    float* h_node = alloc((size_t)M_TOT * HID);            //  8.2 MB
    float* h1c    = alloc((size_t)CHUNK * T_LEN * HID);    // 39.3 MB
    float* xp1c   = alloc((size_t)CHUNK * T_LEN * G3);     // 118  MB
    float* xh     = alloc((size_t)M_TOT * 512);            // 32.8 MB
    float* a_s    = alloc((size_t)M_TOT * 4);
    float* a_d    = alloc((size_t)M_TOT * 4);
    float* segm   = alloc((size_t)M_TOT * 4);
    float* segs   = alloc((size_t)M_TOT * 4);
    float* exbuf  = alloc((size_t)E_TOT * 4);              //  4.1 MB
    float* agg    = alloc((size_t)M_TOT * HID);            //  8.2 MB

    // ---- stacked GRU, 5 chunks of 3200 rows (chunk scratch stays ~L2) ----
    for (int c = 0; c < M_TOT / CHUNK; ++c) {
        int chunk0 = c * CHUNK;
        gru0_scan<<<CHUNK / 64, 128, 0, stream>>>(x, Wih0, Whh0, bih0, bhh0, h1c, chunk0);
        gemm_h128<24, 4, G3, true><<<(CHUNK * T_LEN) / 64, 128, 0, stream>>>(h1c, Wih1, bih1, xp1c);
        gru1_scan<<<CHUNK / 64, 128, 0, stream>>>(xp1c, Whh1, bhh1, h_node, chunk0);
    }

    // ---- two GAT + residual + LayerNorm rounds ----
    for (int r = 0; r < 2; ++r) {
        gat_init<<<(M_TOT * HID + 255) / 256, 256, 0, stream>>>(segm, segs, agg);
        gemm_h128<16, 2, 512, false><<<M_TOT / 32, 128, 0, stream>>>(h_node, gW[r], nullptr, xh);
        att_kernel<<<(M_TOT * 4) / 256, 256, 0, stream>>>(xh, gAsrc[r], gAdst[r], a_s, a_d);
        edge_max<<<E_TOT / 256, 256, 0, stream>>>(ei, a_s, a_d, segm);
        edge_exp<<<E_TOT / 256, 256, 0, stream>>>(ei, a_s, a_d, segm, segs, exbuf);
        edge_agg<<<E_TOT / 8, 256, 0, stream>>>(ei, exbuf, segs, xh, agg);
        relu_res_ln<<<M_TOT / 8, 256, 0, stream>>>(agg, gB[r], lnG[r], lnB[r], h_node);
    }

    // ---- final projection ----
    fc_kernel<<<(M_TOT * 6 + 255) / 256, 256, 0, stream>>>(h_node, fcW, fcB, out);
}